// Decoder_20486994002617
// MI455X (gfx1250) — compile-verified
//
#include <hip/hip_runtime.h>
#include <hip/hip_bf16.h>
#include <math.h>

// ---------------- model constants ----------------
#define Hh   4096
#define HDd  128
#define NHh  32
#define NKVv 8
#define NREPp 4
#define Ll   2
#define Vv   32000
#define Ss   1024
#define Ii   11008
#define EPSf 1e-6f

typedef float v2f __attribute__((ext_vector_type(2)));
typedef float v8f __attribute__((ext_vector_type(8)));

// =====================================================================
// fp32 GEMM: C[M,N] = (beta ? C : 0) + A[M,K] * op(B)[K,N]
// V_WMMA_F32_16X16X4_F32, wave32.
// Block = 256 threads = 8 waves (2 in M x 4 in N).
// Wave tile = 32x32 (2x2 register blocking, 4 v8f accumulators).
// Block tile = 64(M) x 128(N); K-chunk = 16 staged through LDS for A with
// a one-deep register prefetch pipeline (next chunk's global load issued
// before the current chunk's 16 WMMAs).
// TRANSB is a template parameter so each instantiation has a branch-free
// inner loop. Requires M%64==0, N%128==0, K%16==0 (true for all uses).
//
// Fragment layouts (CDNA5 ISA 7.12.2, wave32):
//   A 16x4 f32 : lane r=lane&15 -> row M=r ; vgpr0 = K=hi*2, vgpr1 = K=hi*2+1
//   B 4x16 f32 : vgpr0 = rows K={0(l0-15),2(l16-31)}, vgpr1 = rows K={1,3}, col N=lane&15
//   C/D 16x16  : vgpr j -> row M = j + hi*8, col N = lane&15
// =====================================================================
#define A_STRIDE 18   // 16 k-floats + 2 pad: keeps b64 LDS reads 8B-aligned, spreads banks

template <int TRANSB>
__global__ __launch_bounds__(256)
void k_gemm_f32(const float* __restrict__ A, int lda,
                const float* __restrict__ B, int ldb,
                float* __restrict__ C, int ldc,
                int M, int N, int K, int beta)
{
    __shared__ float As[64 * A_STRIDE];

    const int tid  = threadIdx.x;
    const int lane = tid & 31;
    const int wave = tid >> 5;
    const int wm = wave >> 2;          // 0..1
    const int wn = wave & 3;           // 0..3
    const int mblk = blockIdx.y * 64;
    const int nblk = blockIdx.x * 128;
    const int m0 = mblk + wm * 32;
    const int n0 = nblk + wn * 32;
    const int r  = lane & 15;
    const int hi = lane >> 4;

    v8f acc00 = {}, acc01 = {}, acc10 = {}, acc11 = {};
    if (beta) {
#pragma unroll
        for (int j = 0; j < 8; ++j) {
            const size_t row0 = (size_t)(m0 + j + hi * 8) * ldc;
            const size_t row1 = (size_t)(m0 + 16 + j + hi * 8) * ldc;
            acc00[j] = C[row0 + n0 + r];
            acc01[j] = C[row0 + n0 + 16 + r];
            acc10[j] = C[row1 + n0 + r];
            acc11[j] = C[row1 + n0 + 16 + r];
        }
    }

    // A staging: thread t loads row (t>>2), 4 consecutive k-floats at ((t&3)*4)
    const int ar = tid >> 2;
    const int ac = (tid & 3) * 4;
    const float* Ag = A + (size_t)(mblk + ar) * lda + ac;
    float* asw = &As[ar * A_STRIDE + ac];
    const float* a_lds0 = &As[(wm * 32 + r) * A_STRIDE];
    const float* a_lds1 = &As[(wm * 32 + 16 + r) * A_STRIDE];

    float4 av = *(const float4*)(Ag);  // prefetch chunk 0

    for (int k0 = 0; k0 < K; k0 += 16) {
        asw[0] = av.x; asw[1] = av.y; asw[2] = av.z; asw[3] = av.w;
        __syncthreads();
        if (k0 + 16 < K)               // prefetch next chunk under this chunk's math
            av = *(const float4*)(Ag + k0 + 16);

#pragma unroll
        for (int kk = 0; kk < 16; kk += 4) {
            const int kf = kk + hi * 2;
            v2f a0, a1, b0, b1;
            {
                const float2 f0 = *(const float2*)(a_lds0 + kf);
                const float2 f1 = *(const float2*)(a_lds1 + kf);
                a0.x = f0.x; a0.y = f0.y;
                a1.x = f1.x; a1.y = f1.y;
            }
            if (TRANSB) {
                const float* Bp0 = B + (size_t)(n0 + r) * ldb + k0 + kf;
                const float* Bp1 = B + (size_t)(n0 + 16 + r) * ldb + k0 + kf;
                b0.x = Bp0[0]; b0.y = Bp0[1];
                b1.x = Bp1[0]; b1.y = Bp1[1];
            } else {
                const float* Bp = B + (size_t)(k0 + kf) * ldb;
                b0.x = Bp[n0 + r];      b0.y = Bp[(size_t)ldb + n0 + r];
                b1.x = Bp[n0 + 16 + r]; b1.y = Bp[(size_t)ldb + n0 + 16 + r];
            }
            acc00 = __builtin_amdgcn_wmma_f32_16x16x4_f32(false, a0, false, b0, (short)0, acc00, false, false);
            acc01 = __builtin_amdgcn_wmma_f32_16x16x4_f32(false, a0, false, b1, (short)0, acc01, false, false);
            acc10 = __builtin_amdgcn_wmma_f32_16x16x4_f32(false, a1, false, b0, (short)0, acc10, false, false);
            acc11 = __builtin_amdgcn_wmma_f32_16x16x4_f32(false, a1, false, b1, (short)0, acc11, false, false);
        }
        __syncthreads();
    }

#pragma unroll
    for (int j = 0; j < 8; ++j) {
        const size_t row0 = (size_t)(m0 + j + hi * 8) * ldc;
        const size_t row1 = (size_t)(m0 + 16 + j + hi * 8) * ldc;
        C[row0 + n0 + r]      = acc00[j];
        C[row0 + n0 + 16 + r] = acc01[j];
        C[row1 + n0 + r]      = acc10[j];
        C[row1 + n0 + 16 + r] = acc11[j];
    }
}

// ---------------- embedding gather ----------------
__global__ __launch_bounds__(256)
void k_embed(const float* __restrict__ embed, const int* __restrict__ ids,
             float* __restrict__ h)
{
    const int s = blockIdx.x;
    const float* src = embed + (size_t)ids[s] * Hh;
    float* dst = h + (size_t)s * Hh;
    for (int j = threadIdx.x; j < Hh; j += 256) dst[j] = src[j];
}

// ---------------- conv logit -> mask (layer 0 only) ----------------
__global__ __launch_bounds__(256)
void k_conv_mask(const float* __restrict__ h, const float* __restrict__ cw,
                 const float* __restrict__ cb, int* __restrict__ mask)
{
    const int s = blockIdx.x; // 0..S-2
    __shared__ float red[256];
    float a = 0.f;
    for (int j = threadIdx.x; j < Hh; j += 256)
        a += h[(size_t)s * Hh + j] * cw[j] + h[(size_t)(s + 1) * Hh + j] * cw[Hh + j];
    red[threadIdx.x] = a; __syncthreads();
    for (int off = 128; off > 0; off >>= 1) {
        if (threadIdx.x < off) red[threadIdx.x] += red[threadIdx.x + off];
        __syncthreads();
    }
    if (threadIdx.x == 0) mask[s] = (red[0] + cb[0]) > 0.f ? 1 : 0;
}

// ---------------- stable partition (argsort(~keep)) via block scan ----------------
__global__ __launch_bounds__(1024)
void k_prune_scan(const int* __restrict__ mask, const int* __restrict__ labels,
                  int* __restrict__ perm, int* __restrict__ lab2, int* __restrict__ vlen)
{
    __shared__ int sk[Ss];
    const int i = threadIdx.x;
    const int keepi = (i == 0) ? 1 : (mask[i - 1] ? 0 : 1);
    sk[i] = keepi; __syncthreads();
    for (int off = 1; off < Ss; off <<= 1) {
        int v = (i >= off) ? sk[i - off] : 0;
        __syncthreads();
        sk[i] += v;
        __syncthreads();
    }
    const int total = sk[Ss - 1];
    const int excl  = sk[i] - keepi;
    const int dest  = keepi ? excl : (total + (i - excl));
    perm[dest] = i;
    lab2[dest] = labels[i];
    if (i == 0) vlen[0] = total;
}

// ---------------- merge + reorder rows ----------------
__global__ __launch_bounds__(256)
void k_prune_gather(const float* __restrict__ hin, float* __restrict__ hout,
                    const int* __restrict__ perm, const int* __restrict__ mask)
{
    const int i = blockIdx.x;
    const int src = perm[i];
    const bool mg = (src < Ss - 1) && mask[src];
    for (int j = threadIdx.x; j < Hh; j += 256) {
        float v = hin[(size_t)src * Hh + j];
        if (mg) v = 0.5f * (v + hin[(size_t)(src + 1) * Hh + j]);
        hout[(size_t)i * Hh + j] = v;
    }
}

// ---------------- RMSNorm ----------------
__global__ __launch_bounds__(256)
void k_rmsnorm(const float* __restrict__ h, const float* __restrict__ w,
               float* __restrict__ out)
{
    const int s = blockIdx.x;
    const float* p = h + (size_t)s * Hh;
    __shared__ float red[256];
    float a = 0.f;
    for (int j = threadIdx.x; j < Hh; j += 256) { float v = p[j]; a += v * v; }
    red[threadIdx.x] = a; __syncthreads();
    for (int off = 128; off > 0; off >>= 1) {
        if (threadIdx.x < off) red[threadIdx.x] += red[threadIdx.x + off];
        __syncthreads();
    }
    const float inv = rsqrtf(red[0] / (float)Hh + EPSf);
    for (int j = threadIdx.x; j < Hh; j += 256)
        out[(size_t)s * Hh + j] = p[j] * inv * w[j];
}

// ---------------- RoPE (in place; pair d, d+64 share cos/sin) ----------------
__global__ __launch_bounds__(256)
void k_rope(float* __restrict__ t, int nheads)
{
    const int s = blockIdx.x;
    float* row = t + (size_t)s * nheads * HDd;
    for (int idx = threadIdx.x; idx < nheads * (HDd / 2); idx += 256) {
        const int hh = idx / (HDd / 2);
        const int d  = idx % (HDd / 2);
        const float invf = powf(10000.0f, -(float)d / (float)(HDd / 2));
        const float ang = (float)s * invf;
        const float c = cosf(ang), si = sinf(ang);
        float x1 = row[hh * HDd + d];
        float x2 = row[hh * HDd + d + HDd / 2];
        row[hh * HDd + d]            = x1 * c - x2 * si;
        row[hh * HDd + d + HDd / 2]  = x2 * c + x1 * si;
    }
}

// ---------------- causal softmax over one head's score matrix ----------------
__global__ __launch_bounds__(256)
void k_softmax_causal(float* __restrict__ sc, float scale)
{
    const int row = blockIdx.x;
    float* p = sc + (size_t)row * Ss;
    const int n = row + 1;
    __shared__ float red[256];
    float m = -3.0e38f;
    for (int j = threadIdx.x; j < n; j += 256) m = fmaxf(m, p[j] * scale);
    red[threadIdx.x] = m; __syncthreads();
    for (int off = 128; off > 0; off >>= 1) {
        if (threadIdx.x < off) red[threadIdx.x] = fmaxf(red[threadIdx.x], red[threadIdx.x + off]);
        __syncthreads();
    }
    m = red[0]; __syncthreads();
    float ssum = 0.f;
    for (int j = threadIdx.x; j < n; j += 256) {
        float e = expf(p[j] * scale - m);
        p[j] = e; ssum += e;
    }
    red[threadIdx.x] = ssum; __syncthreads();
    for (int off = 128; off > 0; off >>= 1) {
        if (threadIdx.x < off) red[threadIdx.x] += red[threadIdx.x + off];
        __syncthreads();
    }
    const float rs = 1.0f / red[0];
    for (int j = threadIdx.x; j < Ss; j += 256)
        p[j] = (j < n) ? p[j] * rs : 0.0f;
}

// ---------------- silu(gate) * up, in place into gate ----------------
__global__ __launch_bounds__(256)
void k_silu_mul(float* __restrict__ g, const float* __restrict__ u, int n)
{
    const int i = blockIdx.x * 256 + threadIdx.x;
    if (i < n) { float x = g[i]; g[i] = (x / (1.0f + expf(-x))) * u[i]; }
}

// ---------------- cross-entropy per row ----------------
__global__ __launch_bounds__(256)
void k_ce(const float* __restrict__ logits, const int* __restrict__ lab2,
          const int* __restrict__ vlen, float* __restrict__ cew)
{
    const int s = blockIdx.x; // 0..S-2
    const float* p = logits + (size_t)s * Vv;
    __shared__ float red[256];
    float m = -3.0e38f;
    for (int j = threadIdx.x; j < Vv; j += 256) m = fmaxf(m, p[j]);
    red[threadIdx.x] = m; __syncthreads();
    for (int off = 128; off > 0; off >>= 1) {
        if (threadIdx.x < off) red[threadIdx.x] = fmaxf(red[threadIdx.x], red[threadIdx.x + off]);
        __syncthreads();
    }
    m = red[0]; __syncthreads();
    float ssum = 0.f;
    for (int j = threadIdx.x; j < Vv; j += 256) ssum += expf(p[j] - m);
    red[threadIdx.x] = ssum; __syncthreads();
    for (int off = 128; off > 0; off >>= 1) {
        if (threadIdx.x < off) red[threadIdx.x] += red[threadIdx.x + off];
        __syncthreads();
    }
    if (threadIdx.x == 0) {
        const int tgt = lab2[s + 1];
        const float lp = p[tgt] - m - logf(red[0]);
        cew[s] = (s < vlen[0] - 1) ? -lp : 0.0f;
    }
}

// ---------------- final reduction ----------------
__global__ __launch_bounds__(256)
void k_loss(const float* __restrict__ cew, const int* __restrict__ vlen,
            float* __restrict__ out)
{
    __shared__ float red[256];
    float a = 0.f;
    for (int j = threadIdx.x; j < Ss - 1; j += 256) a += cew[j];
    red[threadIdx.x] = a; __syncthreads();
    for (int off = 128; off > 0; off >>= 1) {
        if (threadIdx.x < off) red[threadIdx.x] += red[threadIdx.x + off];
        __syncthreads();
    }
    if (threadIdx.x == 0) out[0] = red[0] / (float)(vlen[0] - 1);
}

// =====================================================================
static inline void gemm(const float* A, int lda, const float* B, int ldb, int transB,
                        float* C, int ldc, int M, int N, int K, int beta, hipStream_t s)
{
    dim3 g(N / 128, M / 64);
    if (transB)
        k_gemm_f32<1><<<g, dim3(256), 0, s>>>(A, lda, B, ldb, C, ldc, M, N, K, beta);
    else
        k_gemm_f32<0><<<g, dim3(256), 0, s>>>(A, lda, B, ldb, C, ldc, M, N, K, beta);
}

extern "C" void kernel_launch(void* const* d_in, const int* in_sizes, int n_in,
                              void* d_out, int out_size, void* d_ws, size_t ws_size,
                              hipStream_t stream)
{
    (void)in_sizes; (void)n_in; (void)out_size; (void)ws_size;
    const int*   ids     = (const int*)d_in[0];
    const int*   labels  = (const int*)d_in[1];
    const float* embed   = (const float*)d_in[2];
    const float* conv_w  = (const float*)d_in[3];
    const float* conv_b  = (const float*)d_in[4];
    const float* q_w     = (const float*)d_in[5];
    const float* k_w     = (const float*)d_in[6];
    const float* v_w     = (const float*)d_in[7];
    const float* o_w     = (const float*)d_in[8];
    const float* ln1_w   = (const float*)d_in[9];
    const float* ln2_w   = (const float*)d_in[10];
    const float* gate_w  = (const float*)d_in[11];
    const float* up_w    = (const float*)d_in[12];
    const float* down_w  = (const float*)d_in[13];
    const float* norm_w  = (const float*)d_in[14];
    const float* lm_head = (const float*)d_in[15];
    float* out = (float*)d_out;

    // ---- workspace layout (bytes) ----
    uint8_t* w = (uint8_t*)d_ws;
    const size_t SH  = (size_t)Ss * Hh  * 4;          // 16 MB
    const size_t SKV = (size_t)Ss * NKVv * HDd * 4;   // 4 MB
    size_t off = 0;
    float* h0   = (float*)(w + off); off += SH;
    float* h1   = (float*)(w + off); off += SH;
    float* x    = (float*)(w + off); off += SH;
    int*   mask = (int*)(w + off);   off += (size_t)Ss * 4;
    int*   perm = (int*)(w + off);   off += (size_t)Ss * 4;
    int*   lab2 = (int*)(w + off);   off += (size_t)Ss * 4;
    int*   vlen = (int*)(w + off);   off += 256;           // padded
    float* cew  = (float*)(w + off); off += (size_t)Ss * 4;
    off = (off + 255) & ~(size_t)255;
    const size_t off_q = off;
    float* q    = (float*)(w + off); off += SH;
    float* kb   = (float*)(w + off); off += SKV;
    float* vb   = (float*)(w + off); off += SKV;
    float* ao   = (float*)(w + off); off += SH;
    float* sc   = (float*)(w + off); off += (size_t)Ss * Ss * 4;
    float* gate = (float*)(w + off); off += (size_t)Ss * Ii * 4;
    float* up   = (float*)(w + off); off += (size_t)Ss * Ii * 4;
    // logits (S*V f32 = 131 MB) alias the q..up region (136 MB), dead by then
    float* logits = (float*)(w + off_q);

    const float scale = 0.08838834764831845f; // 1/sqrt(128)

    // ---- embed + prune (layer 0) ----
    k_embed<<<Ss, 256, 0, stream>>>(embed, ids, h0);
    k_conv_mask<<<Ss - 1, 256, 0, stream>>>(h0, conv_w, conv_b, mask);
    k_prune_scan<<<1, 1024, 0, stream>>>(mask, labels, perm, lab2, vlen);
    k_prune_gather<<<Ss, 256, 0, stream>>>(h0, h1, perm, mask);
    float* h = h1;

    // ---- transformer layers ----
    for (int l = 0; l < Ll; ++l) {
        k_rmsnorm<<<Ss, 256, 0, stream>>>(h, ln1_w + (size_t)l * Hh, x);
        gemm(x, Hh, q_w + (size_t)l * Hh * (NHh * HDd), NHh * HDd, 0,
             q, NHh * HDd, Ss, NHh * HDd, Hh, 0, stream);
        gemm(x, Hh, k_w + (size_t)l * Hh * (NKVv * HDd), NKVv * HDd, 0,
             kb, NKVv * HDd, Ss, NKVv * HDd, Hh, 0, stream);
        gemm(x, Hh, v_w + (size_t)l * Hh * (NKVv * HDd), NKVv * HDd, 0,
             vb, NKVv * HDd, Ss, NKVv * HDd, Hh, 0, stream);
        k_rope<<<Ss, 256, 0, stream>>>(q, NHh);
        k_rope<<<Ss, 256, 0, stream>>>(kb, NKVv);

        for (int hh = 0; hh < NHh; ++hh) {
            const int kvh = hh / NREPp;
            // scores = q_h @ k_h^T   (A: S x HD, B^T: S x HD)
            gemm(q + hh * HDd, NHh * HDd,
                 kb + kvh * HDd, NKVv * HDd, 1,
                 sc, Ss, Ss, Ss, HDd, 0, stream);
            k_softmax_causal<<<Ss, 256, 0, stream>>>(sc, scale);
            // o_h = attn @ v_h
            gemm(sc, Ss,
                 vb + kvh * HDd, NKVv * HDd, 0,
                 ao + hh * HDd, NHh * HDd, Ss, HDd, Ss, 0, stream);
        }
        // h += ao @ o_w
        gemm(ao, NHh * HDd, o_w + (size_t)l * (NHh * HDd) * Hh, Hh, 0,
             h, Hh, Ss, Hh, NHh * HDd, 1, stream);

        // MLP
        k_rmsnorm<<<Ss, 256, 0, stream>>>(h, ln2_w + (size_t)l * Hh, x);
        gemm(x, Hh, gate_w + (size_t)l * Hh * Ii, Ii, 0, gate, Ii, Ss, Ii, Hh, 0, stream);
        gemm(x, Hh, up_w   + (size_t)l * Hh * Ii, Ii, 0, up,   Ii, Ss, Ii, Hh, 0, stream);
        {
            const int n = Ss * Ii;
            k_silu_mul<<<(n + 255) / 256, 256, 0, stream>>>(gate, up, n);
        }
        gemm(gate, Ii, down_w + (size_t)l * Ii * Hh, Hh, 0, h, Hh, Ss, Hh, Ii, 1, stream);
    }

    // ---- head + loss ----
    k_rmsnorm<<<Ss, 256, 0, stream>>>(h, norm_w, x);
    gemm(x, Hh, lm_head, Vv, 0, logits, Vv, Ss, Vv, Hh, 0, stream);
    k_ce<<<Ss - 1, 256, 0, stream>>>(logits, lab2, vlen, cew);
    k_loss<<<1, 256, 0, stream>>>(cew, vlen, out);
}